// HM_54348516163720
// MI455X (gfx1250) — compile-verified
//
#include <hip/hip_runtime.h>

typedef unsigned int u32;

#define RADIX  256
#define RBITS  8
#define NTHR   256          // threads per block (8 waves of 32)
#define EPT    16           // elements per thread
#define TILE   4096         // NTHR * EPT; must equal numBlocks (N = 2^24)

// ---------- order-preserving float <-> uint key transform (pure bit ops) ----------
__device__ __forceinline__ u32 keyFwdU(u32 u) {
    u32 mask = (u & 0x80000000u) ? 0xFFFFFFFFu : 0x80000000u;
    return u ^ mask;
}
__device__ __forceinline__ float keyInv(u32 k) {
    u32 mask = (k & 0x80000000u) ? 0x80000000u : 0xFFFFFFFFu;
    return __uint_as_float(k ^ mask);
}

// ---------- per-block digit histogram (b128 loads), digit-major output ----------
template <bool FROM_FLOAT>
__global__ void hm_radix_hist(const u32* __restrict__ src, u32* __restrict__ counts,
                              int shift, int numBlocks) {
    __shared__ u32 hist[RADIX];
    const int t = threadIdx.x, b = blockIdx.x;
    hist[t] = 0;
    __syncthreads();
    const uint4* p = (const uint4*)(src + (size_t)b * TILE);
#pragma unroll
    for (int j = 0; j < TILE / 4 / NTHR; ++j) {       // 4 iterations of uint4
        uint4 q = p[j * NTHR + t];
        if (FROM_FLOAT) { q.x = keyFwdU(q.x); q.y = keyFwdU(q.y);
                          q.z = keyFwdU(q.z); q.w = keyFwdU(q.w); }
        atomicAdd(&hist[(q.x >> shift) & (RADIX - 1)], 1u);
        atomicAdd(&hist[(q.y >> shift) & (RADIX - 1)], 1u);
        atomicAdd(&hist[(q.z >> shift) & (RADIX - 1)], 1u);
        atomicAdd(&hist[(q.w >> shift) & (RADIX - 1)], 1u);
    }
    __syncthreads();
    counts[(size_t)t * numBlocks + b] = hist[t];
}

// ---------- exclusive scan over the 1M counters (per-digit segments) ----------
__global__ void hm_scan_blocks(u32* __restrict__ data, u32* __restrict__ blockSums) {
    __shared__ u32 lds[NTHR];
    const int t = threadIdx.x, b = blockIdx.x;
    u32* p = data + (size_t)b * TILE + t * EPT;
    u32 v[EPT];
    u32 s = 0;
#pragma unroll
    for (int k = 0; k < EPT; ++k) { v[k] = p[k]; s += v[k]; }
    lds[t] = s;
    __syncthreads();
    for (int off = 1; off < NTHR; off <<= 1) {
        u32 add = (t >= off) ? lds[t - off] : 0u;
        __syncthreads();
        lds[t] += add;
        __syncthreads();
    }
    u32 inc = lds[t];
    u32 run = inc - s;
#pragma unroll
    for (int k = 0; k < EPT; ++k) { u32 tmp = v[k]; p[k] = run; run += tmp; }
    if (t == NTHR - 1) blockSums[b] = inc;
}

__global__ void hm_scan_sums(u32* __restrict__ sums) {   // exactly NTHR entries, 1 block
    __shared__ u32 lds[NTHR];
    const int t = threadIdx.x;
    u32 v = sums[t];
    lds[t] = v;
    __syncthreads();
    for (int off = 1; off < NTHR; off <<= 1) {
        u32 add = (t >= off) ? lds[t - off] : 0u;
        __syncthreads();
        lds[t] += add;
        __syncthreads();
    }
    sums[t] = lds[t] - v;      // exclusive per-digit base
}

// ---------- stable scatter: ballot multisplit + LDS staging for coalesced stores ----------
template <bool FROM_FLOAT>
__global__ void hm_radix_scatter(const u32* __restrict__ src, u32* __restrict__ dst,
                                 const u32* __restrict__ counts, const u32* __restrict__ bsumsEx,
                                 int shift, int numBlocks) {
    __shared__ u32 digitStart[RADIX];    // global base for (digit, this block)
    __shared__ u32 running[RADIX];       // per-digit emitted-so-far in this tile
    __shared__ u32 localStart[RADIX];    // per-digit start within tile
    __shared__ u32 warpCnt[RADIX * 8];   // per-iteration per-warp digit counts
    __shared__ u32 tileLds[TILE];        // locally digit-sorted tile
    const int t = threadIdx.x, b = blockIdx.x;
    const int lane = t & 31, warp = t >> 5;
    digitStart[t] = counts[(size_t)t * numBlocks + b] + bsumsEx[t];  // fused scan_add
    running[t] = 0;
    const u32* p = src + (size_t)b * TILE;
    u32 vv[EPT], rk[EPT];
#pragma unroll
    for (int j = 0; j < EPT; ++j) {
        __syncthreads();                      // init / previous-iteration reads done
#pragma unroll
        for (int w = 0; w < 8; ++w) warpCnt[t * 8 + w] = 0;
        __syncthreads();
        u32 v = p[j * NTHR + t];
        if (FROM_FLOAT) v = keyFwdU(v);
        const u32 d = (v >> shift) & (RADIX - 1);
        u32 peers = 0xFFFFFFFFu;              // lanes in this wave with same digit
#pragma unroll
        for (int bit = 0; bit < RBITS; ++bit) {
            bool pr = (d >> bit) & 1u;
            u32 bal = (u32)__ballot(pr);      // wave32: low 32 bits valid
            peers &= pr ? bal : ~bal;
        }
        const u32 rankInWarp = __popc(peers & ((1u << lane) - 1u));
        if (rankInWarp == 0) warpCnt[d * 8 + warp] = (u32)__popc(peers);
        __syncthreads();
        u32 wp = 0;
        for (int w = 0; w < warp; ++w) wp += warpCnt[d * 8 + w];
        u32 tot = 0;
#pragma unroll
        for (int w = 0; w < 8; ++w) tot += warpCnt[t * 8 + w];
        vv[j] = v;
        rk[j] = running[d] + wp + rankInWarp; // stable local (within-digit) rank
        __syncthreads();
        running[t] += tot;
    }
    __syncthreads();
    // exclusive scan of tile digit counts -> localStart
    const u32 cnt = running[t];
    localStart[t] = cnt;
    __syncthreads();
    for (int off = 1; off < RADIX; off <<= 1) {
        u32 add = (t >= off) ? localStart[t - off] : 0u;
        __syncthreads();
        localStart[t] += add;
        __syncthreads();
    }
    const u32 excl = localStart[t] - cnt;
    __syncthreads();
    localStart[t] = excl;
    __syncthreads();
    // local counting sort into LDS
#pragma unroll
    for (int j = 0; j < EPT; ++j) {
        const u32 d = (vv[j] >> shift) & (RADIX - 1);
        tileLds[localStart[d] + rk[j]] = vv[j];
    }
    __syncthreads();
    // near-coalesced global scatter (breaks only at digit boundaries)
#pragma unroll
    for (int j = 0; j < EPT; ++j) {
        const int k = j * NTHR + t;
        const u32 v = tileLds[k];
        const u32 d = (v >> shift) & (RADIX - 1);
        dst[digitStart[d] + ((u32)k - localStart[d])] = v;
    }
}

// ---------- final: rank content by binary search, gather sorted style ----------
__global__ void hm_match(const float* __restrict__ content,
                         const u32* __restrict__ sortedC,
                         const u32* __restrict__ sortedS,
                         float* __restrict__ out, int n) {
    int i = blockIdx.x * blockDim.x + threadIdx.x;
    if (i >= n) return;
    const u32 key = keyFwdU(__float_as_uint(content[i]));
    u32 lo = 0, hi = (u32)n;
    while (lo < hi) {
        u32 mid = (lo + hi) >> 1;
        if (sortedC[mid] < key) lo = mid + 1; else hi = mid;
    }
    if (lo >= (u32)n) lo = (u32)n - 1;   // safety (key always present)
    out[i] = keyInv(sortedS[lo]);
}

extern "C" void kernel_launch(void* const* d_in, const int* in_sizes, int n_in,
                              void* d_out, int out_size, void* d_ws, size_t ws_size,
                              hipStream_t stream) {
    const float* content = (const float*)d_in[0];
    const float* style   = (const float*)d_in[1];
    const int n = in_sizes[0];                 // 16*64*128*128 = 2^24
    const int numBlocks  = n / TILE;           // 4096 (== TILE, required for fused scan_add)
    const int M          = RADIX * numBlocks;  // 1<<20 counters
    const int scanBlocks = M / TILE;           // 256 == NTHR (required by hm_scan_sums)

    u32* S0     = (u32*)d_ws;
    u32* S1     = S0 + n;
    u32* counts = S1 + n;
    u32* bsums  = counts + M;
    u32* OUT    = (u32*)d_out;                 // ping-pong scratch until final gather
    float* fout = (float*)d_out;

    auto scanCounts = [&]() {
        hm_scan_blocks<<<scanBlocks, NTHR, 0, stream>>>(counts, bsums);
        hm_scan_sums  <<<1,          NTHR, 0, stream>>>(bsums);
    };
    auto passF = [&](const float* srcF, u32* dst) {        // pass 0: floats -> keys
        const u32* src = (const u32*)srcF;
        hm_radix_hist<true>   <<<numBlocks, NTHR, 0, stream>>>(src, counts, 0, numBlocks);
        scanCounts();
        hm_radix_scatter<true><<<numBlocks, NTHR, 0, stream>>>(src, dst, counts, bsums, 0, numBlocks);
    };
    auto passK = [&](const u32* src, u32* dst, int shift) { // passes 1..3: keys
        hm_radix_hist<false>   <<<numBlocks, NTHR, 0, stream>>>(src, counts, shift, numBlocks);
        scanCounts();
        hm_radix_scatter<false><<<numBlocks, NTHR, 0, stream>>>(src, dst, counts, bsums, shift, numBlocks);
    };

    // ---- style: F -> S0 -> S1 -> S0 -> S1 (sorted style keys in S1) ----
    passF(style, S0);
    passK(S0, S1, 8);
    passK(S1, S0, 16);
    passK(S0, S1, 24);

    // ---- content: F -> OUT -> S0 -> OUT -> S0 (sorted content keys in S0) ----
    passF(content, OUT);
    passK(OUT, S0, 8);
    passK(S0, OUT, 16);
    passK(OUT, S0, 24);

    // ---- rank + gather (d_out now free to receive final floats) ----
    hm_match<<<(n + NTHR - 1) / NTHR, NTHR, 0, stream>>>(content, S0, S1, fout, n);
}